// ClusterActivation_33260226740919
// MI455X (gfx1250) — compile-verified
//
#include <hip/hip_runtime.h>

typedef float v2f __attribute__((ext_vector_type(2)));
typedef float v4f __attribute__((ext_vector_type(4)));
typedef float v8f __attribute__((ext_vector_type(8)));
typedef int   v4i_vs __attribute__((vector_size(16)));   // matches builtin param type

#define DIM   512
#define NC    4
#define ROWS  16
#define PADF  516  // floats per padded LDS row: 16B-aligned, bank-skewed (516%64==4)

// ---- CDNA5 async global->LDS staging (ASYNCcnt-tracked), with safe fallback ----
#if defined(__has_builtin)
#if __has_builtin(__builtin_amdgcn_global_load_async_to_lds_b128) && \
    __has_builtin(__builtin_amdgcn_s_wait_asynccnt)
#define USE_ASYNC_LDS 1
#endif
#endif
#ifndef USE_ASYNC_LDS
#define USE_ASYNC_LDS 0
#endif

__device__ __forceinline__ float wave_reduce_add(float v) {
#pragma unroll
    for (int off = 16; off > 0; off >>= 1)
        v += __shfl_xor(v, off, 32);
    return v;
}

__device__ __forceinline__ float gelu_tanh_f(float x) {
    const float c0 = 0.7978845608028654f;  // sqrt(2/pi)
    const float c1 = 0.044715f;
    float x3 = x * x * x;
    float t  = tanhf(c0 * (x + c1 * x3));
    return 0.5f * x * (1.0f + t);
}

__device__ __forceinline__ float apply_act(int lab, float x) {
    switch (lab) {
        case 0:  return fmaxf(x, 0.0f);           // relu
        case 1:  return tanhf(x);                 // tanh
        case 2:  return gelu_tanh_f(x);           // gelu (tanh approx = jax default)
        default: return 1.0f / (1.0f + expf(-x)); // sigmoid
    }
}

__global__ __launch_bounds__(32) void
ClusterActivation_33260226740919_kernel(const float* __restrict__ x,
                                        const float* __restrict__ cent,
                                        float* __restrict__ out, int N) {
    __shared__ float xs[ROWS * PADF];
    __shared__ float cs[NC * PADF];
    __shared__ float meanL[ROWS];
    __shared__ float rstdL[ROWS];
    __shared__ int   labelL[ROWS];

    const int lane = threadIdx.x;   // 0..31 (wave32)
    const int row0 = blockIdx.x * ROWS;
    const int rsub = lane & 15;     // row-in-tile / cluster-row for this lane
    const int hi   = lane >> 4;     // 0: K-low half, 1: K-high half (WMMA layout)

    // ---- kick off async staging of the 16-row x tile straight into LDS ----
#if USE_ASYNC_LDS
    for (int r = 0; r < ROWS; ++r) {
        int row = row0 + r; if (row >= N) row = N - 1;
        const float* src = x + (size_t)row * DIM;
        float* dstRow = xs + r * PADF;
#pragma unroll
        for (int k4 = 0; k4 < DIM / 4 / 32; ++k4) {
            int k = (k4 * 32 + lane) * 4;  // float index, 16B granules
            __builtin_amdgcn_global_load_async_to_lds_b128(
                (v4i_vs*)(src + k), (v4i_vs*)(dstRow + k), 0, 0);
        }
    }
#endif

    // ---- stage centroids into LDS + per-cluster squared norms ----
    float cn2[NC];
#pragma unroll
    for (int c = 0; c < NC; ++c) {
        float sq = 0.f;
#pragma unroll
        for (int k = lane; k < DIM; k += 32) {
            float v = cent[c * DIM + k];
            cs[c * PADF + k] = v;
            sq += v * v;
        }
        cn2[c] = wave_reduce_add(sq);
    }

#if USE_ASYNC_LDS
    __builtin_amdgcn_s_wait_asynccnt(0);   // x tile resident in LDS
#else
    for (int r = 0; r < ROWS; ++r) {
        int row = row0 + r; if (row >= N) row = N - 1;
        const v4f* __restrict__ src = (const v4f*)(x + (size_t)row * DIM);
        v4f* dst = (v4f*)(xs + r * PADF);
#pragma unroll
        for (int k4 = lane; k4 < DIM / 4; k4 += 32)
            dst[k4] = __builtin_nontemporal_load(&src[k4]);
    }
#endif
    __syncthreads();

    // ---- row stats: 2 lanes per row (halves), merged via shfl_xor(16) ----
    float s = 0.f, ss = 0.f;
    {
        const float* rp = xs + rsub * PADF + hi * (DIM / 2);
#pragma unroll 8
        for (int k = 0; k < DIM / 2; k += 4) {
            v4f v = *(const v4f*)(rp + k);
            s  += v.x + v.y + v.z + v.w;
            ss += v.x * v.x + v.y * v.y + v.z * v.z + v.w * v.w;
        }
        s  += __shfl_xor(s, 16, 32);
        ss += __shfl_xor(ss, 16, 32);
    }
    float mean = s * (1.0f / DIM);
    float var  = (ss - s * mean) * (1.0f / (DIM - 1));  // unbiased (ddof=1)
    float rstd = rsqrtf(var + 1e-5f);
    if (lane < ROWS) { meanL[lane] = mean; rstdL[lane] = rstd; }

    // ---- x @ centroids^T via V_WMMA_F32_16X16X4_F32 (exact fp32) ----
    // A rows 4..15 are duplicates of clusters 0..3 (rsub & 3): no EXEC masking,
    // duplicate D rows are simply never read.
    // Lane layout: lanes 0-15 hold K = {k0,k0+1}, lanes 16-31 hold K = {k0+2,k0+3}.
    v8f acc = {};
    const int koff = hi * 2;
    const float* aBase = cs + (rsub & 3) * PADF + koff;
    const float* bBase = xs + rsub * PADF + koff;
    for (int k0 = 0; k0 < DIM; k0 += 4) {
        v2f a = *(const v2f*)(aBase + k0);
        v2f b = *(const v2f*)(bBase + k0);
        acc = __builtin_amdgcn_wmma_f32_16x16x4_f32(false, a, false, b,
                                                    (short)0, acc, false, false);
    }

    // D layout: lane l (l<16), VGPR j -> dot(cent[j], x[row0+l]), j = 0..3
    {
        float best = cn2[0] - 2.0f * acc[0];
        int bi = 0;
#pragma unroll
        for (int j = 1; j < NC; ++j) {
            float sc = cn2[j] - 2.0f * acc[j];
            if (sc < best) { best = sc; bi = j; }   // first-min tie-break == argmin
        }
        if (lane < ROWS) labelL[lane] = bi;
    }
    __syncthreads();

    // ---- normalize + per-cluster activation + coalesced NT b128 store ----
    for (int r = 0; r < ROWS; ++r) {
        int row = row0 + r;
        if (row >= N) break;
        float m  = meanL[r];
        float rs = rstdL[r];
        int  lab = labelL[r];   // wave-uniform -> non-divergent switch
        const v4f* src = (const v4f*)(xs + r * PADF);
        v4f* dst = (v4f*)(out + (size_t)row * DIM);
#pragma unroll
        for (int k4 = lane; k4 < DIM / 4; k4 += 32) {
            v4f v = src[k4];
            v4f o;
            o.x = apply_act(lab, (v.x - m) * rs);
            o.y = apply_act(lab, (v.y - m) * rs);
            o.z = apply_act(lab, (v.z - m) * rs);
            o.w = apply_act(lab, (v.w - m) * rs);
            __builtin_nontemporal_store(o, &dst[k4]);
        }
    }
}

extern "C" void kernel_launch(void* const* d_in, const int* in_sizes, int n_in,
                              void* d_out, int out_size, void* d_ws, size_t ws_size,
                              hipStream_t stream) {
    const float* x    = (const float*)d_in[0];
    const float* cent = (const float*)d_in[1];
    float* out        = (float*)d_out;
    int N = in_sizes[0] / DIM;             // 100000
    int blocks = (N + ROWS - 1) / ROWS;    // 6250
    ClusterActivation_33260226740919_kernel<<<blocks, 32, 0, stream>>>(x, cent, out, N);
}